// GCN_2052994367627
// MI455X (gfx1250) — compile-verified
//
#include <hip/hip_runtime.h>
#include <cstdint>
#include <cstddef>

// ---------------------------------------------------------------------------
// GCN 2-layer forward for MI455X (gfx1250, wave32).
//   deg/dinv  -> fp32 atomics
//   GEMMs     -> V_WMMA_F32_16X16X4_F32 (exact fp32 tensor op; GEMMs are not
//                the bottleneck, so no precision downcast)
//   aggregate -> per-edge wave, global_atomic_add_f32 (L2-resident, 192MB L2)
// ---------------------------------------------------------------------------

typedef __attribute__((ext_vector_type(2))) float v2f;
typedef __attribute__((ext_vector_type(8))) float v8f;

#define WAVES_PER_BLOCK 8

// ---- degree / normalization -----------------------------------------------

__global__ __launch_bounds__(256) void k_init_deg(float* __restrict__ deg, int n) {
  int i = blockIdx.x * blockDim.x + threadIdx.x;
  if (i < n) deg[i] = 1.0f;  // self-loop contributes 1 to every node
}

__global__ __launch_bounds__(256) void k_count_deg(const int* __restrict__ col,
                                                   float* __restrict__ deg, int e) {
  int i = blockIdx.x * blockDim.x + threadIdx.x;
  if (i < e) atomicAdd(&deg[col[i]], 1.0f);
}

__global__ __launch_bounds__(256) void k_rsqrt(float* __restrict__ d, int n) {
  int i = blockIdx.x * blockDim.x + threadIdx.x;
  if (i < n) d[i] = rsqrtf(d[i]);  // deg >= 1 always (self-loops), no zero case
}

// ---- fp32 WMMA GEMM: H[N x NC] = X[N x 128] @ W[128 x NC] -----------------
// One wave computes one 16x16 tile of H, walking K in steps of 4 with
// v_wmma_f32_16x16x4_f32.  EXEC stays all-ones inside the WMMA region
// (tile-validity branch is wave-uniform).
//
// A fragment (16x4 f32, ISA layout): lanes 0-15 hold row m, K = {k, k+1};
//   lanes 16-31 hold row m, K = {k+2, k+3}  -> one aligned float2 per lane.
// B fragment (4x16): VGPR0 = rows {k, k+2}, VGPR1 = rows {k+1, k+3},
//   N = lane & 15 across each half-wave (mirrors the C/D half-wave split).
// C/D (16x16 f32): VGPR j -> row j (lanes 0-15) / row j+8 (lanes 16-31).

template <int NC>
__global__ __launch_bounds__(256) void k_gemm_wmma(const float* __restrict__ X,
                                                   const float* __restrict__ W,
                                                   float* __restrict__ H,
                                                   int nTiles) {
  const int lane = threadIdx.x & 31;
  const int wave = threadIdx.x >> 5;
  const int tile = blockIdx.x * WAVES_PER_BLOCK + wave;
  if (tile >= nTiles) return;  // wave-uniform
  constexpr int nColTiles = NC / 16;
  const int rowBase = (tile / nColTiles) * 16;
  const int colBase = (tile % nColTiles) * 16;
  const int m  = lane & 15;   // row (A) / col (B,C)
  const int hi = lane >> 4;   // half-wave select

  const float* xrow = X + (size_t)(rowBase + m) * 128 + (size_t)(hi * 2);
  const float* wcol = W + colBase + m;

  v8f acc = {};
#pragma unroll
  for (int k = 0; k < 128; k += 4) {
    const float2 av = *reinterpret_cast<const float2*>(xrow + k);  // 8B aligned
    v2f a = {av.x, av.y};
    const int kb = k + hi * 2;
    v2f b = {wcol[(size_t)kb * NC], wcol[(size_t)(kb + 1) * NC]};
    acc = __builtin_amdgcn_wmma_f32_16x16x4_f32(
        /*neg_a=*/false, a, /*neg_b=*/false, b,
        /*c_mod=*/(short)0, acc, /*reuse_a=*/false, /*reuse_b=*/false);
  }

  float* out = H + (size_t)(rowBase + hi * 8) * NC + colBase + m;
#pragma unroll
  for (int j = 0; j < 8; ++j) out[(size_t)j * NC] = acc[j];
}

// ---- self-loop term + bias: AGG = H * dinv^2 + bias -----------------------

template <int F>
__global__ __launch_bounds__(256) void k_selfloop_bias(const float* __restrict__ H,
                                                       const float* __restrict__ dinv,
                                                       const float* __restrict__ bias,
                                                       float* __restrict__ AGG,
                                                       int total) {
  int i = blockIdx.x * blockDim.x + threadIdx.x;
  if (i >= total) return;
  int v = i / F;
  int f = i & (F - 1);
  float d = dinv[v];
  AGG[i] = fmaf(H[i], d * d, bias[f]);
}

// ---- edge scatter: AGG[col] += H[row] * dinv[row]*dinv[col] ---------------
// One wave per edge; each lane carries F/32 contiguous features.

template <int F>
__global__ __launch_bounds__(256) void k_edge_scatter(const int* __restrict__ row,
                                                      const int* __restrict__ col,
                                                      const float* __restrict__ dinv,
                                                      const float* __restrict__ H,
                                                      float* __restrict__ AGG,
                                                      int E) {
  constexpr int PF = F / 32;
  const int lane = threadIdx.x & 31;
  const int e = (int)((blockIdx.x * blockDim.x + threadIdx.x) >> 5);
  if (e >= E) return;  // wave-uniform
  const int r = row[e];
  const int c = col[e];
  const float s = dinv[r] * dinv[c];
  const float* src = H + (size_t)r * F + lane * PF;
  float* dst = AGG + (size_t)c * F + lane * PF;
  if constexpr (PF == 4) {
    const float4 v = *reinterpret_cast<const float4*>(src);  // 16B aligned
    atomicAdd(dst + 0, v.x * s);
    atomicAdd(dst + 1, v.y * s);
    atomicAdd(dst + 2, v.z * s);
    atomicAdd(dst + 3, v.w * s);
  } else {
    const float2 v = *reinterpret_cast<const float2*>(src);  // 8B aligned
    atomicAdd(dst + 0, v.x * s);
    atomicAdd(dst + 1, v.y * s);
  }
}

__global__ __launch_bounds__(256) void k_relu(float* __restrict__ a, int total) {
  int i = blockIdx.x * blockDim.x + threadIdx.x;
  if (i < total) a[i] = fmaxf(a[i], 0.0f);
}

// ---------------------------------------------------------------------------

extern "C" void kernel_launch(void* const* d_in, const int* in_sizes, int n_in,
                              void* d_out, int out_size, void* d_ws, size_t ws_size,
                              hipStream_t stream) {
  const float* x  = (const float*)d_in[0];
  const int*   ei = (const int*)d_in[1];   // edge_index [2, E] (row-major)
  const float* W1 = (const float*)d_in[2];
  const float* b1 = (const float*)d_in[3];
  const float* W2 = (const float*)d_in[4];
  const float* b2 = (const float*)d_in[5];

  constexpr int F1 = 128, F2 = 64;
  const int N = in_sizes[0] / F1;     // 100000 (divisible by 16)
  const int E = in_sizes[1] / 2;      // 1.6M
  const int* row = ei;
  const int* col = ei + E;

  // workspace layout (~128.5 MB): dinv | h1 | agg1 | h2
  char* ws = (char*)d_ws;
  float* dinv = (float*)ws;
  size_t off = ((size_t)N * sizeof(float) + 255) & ~(size_t)255;
  float* h1   = (float*)(ws + off); off += (size_t)N * F1 * sizeof(float);
  float* agg1 = (float*)(ws + off); off += (size_t)N * F1 * sizeof(float);
  float* h2   = (float*)(ws + off);
  float* out  = (float*)d_out;

  // --- normalization ---
  k_init_deg <<<(N + 255) / 256, 256, 0, stream>>>(dinv, N);
  k_count_deg<<<(E + 255) / 256, 256, 0, stream>>>(col, dinv, E);
  k_rsqrt    <<<(N + 255) / 256, 256, 0, stream>>>(dinv, N);

  const int rowTiles = N / 16;

  // --- layer 1: h1 = x @ W1 ; agg1 = scatter + b1 ; relu ---
  {
    const int nT = rowTiles * (F1 / 16);
    k_gemm_wmma<F1><<<(nT + WAVES_PER_BLOCK - 1) / WAVES_PER_BLOCK, 256, 0, stream>>>(
        x, W1, h1, nT);
    const int tot = N * F1;
    k_selfloop_bias<F1><<<(tot + 255) / 256, 256, 0, stream>>>(h1, dinv, b1, agg1, tot);
    k_edge_scatter<F1><<<(E + WAVES_PER_BLOCK - 1) / WAVES_PER_BLOCK, 256, 0, stream>>>(
        row, col, dinv, h1, agg1, E);
    k_relu<<<(tot + 255) / 256, 256, 0, stream>>>(agg1, tot);
  }

  // --- layer 2: h2 = relu_out @ W2 ; d_out = scatter + b2 ---
  {
    const int nT = rowTiles * (F2 / 16);
    k_gemm_wmma<F2><<<(nT + WAVES_PER_BLOCK - 1) / WAVES_PER_BLOCK, 256, 0, stream>>>(
        agg1, W2, h2, nT);
    const int tot = N * F2;
    k_selfloop_bias<F2><<<(tot + 255) / 256, 256, 0, stream>>>(h2, dinv, b2, out, tot);
    k_edge_scatter<F2><<<(E + WAVES_PER_BLOCK - 1) / WAVES_PER_BLOCK, 256, 0, stream>>>(
        row, col, dinv, h2, out, E);
  }
}